// RoITemporalPool_61263413510368
// MI455X (gfx1250) — compile-verified
//
#include <hip/hip_runtime.h>
#include <stdint.h>

// Problem constants (from reference)
#define NUM_ROIS 128
#define NCH      512
#define T_DIM    64
#define HW       49           // 7*7
#define POOLED   4
#define TSCALE   0.125f
#define FEAT_CSTRIDE (T_DIM * HW)          // 3136
#define FEAT_NSTRIDE (NCH * FEAT_CSTRIDE)  // 1605632
#define OUT_CSTRIDE  (POOLED * HW)         // 196
#define OUT_RSTRIDE  (NCH * OUT_CSTRIDE)   // 100352

// gfx1250 async global->LDS path (device compilation pass only)
#if defined(__gfx1250__) && __has_builtin(__builtin_amdgcn_global_load_async_to_lds_b32)
#define HAVE_ASYNC_LDS 1
#else
#define HAVE_ASYNC_LDS 0
#endif

__device__ __forceinline__ void wait_async0() {
#if __has_builtin(__builtin_amdgcn_s_wait_asynccnt)
  __builtin_amdgcn_s_wait_asynccnt(0);
#else
  asm volatile("s_wait_asynccnt 0" ::: "memory");
#endif
}

__global__ __launch_bounds__(256)
void roi_temporal_pool_kernel(const float* __restrict__ features,
                              const float* __restrict__ rois,
                              float* __restrict__ out) {
  // Staging buffer for the union temporal window of this (roi, channel):
  // worst case T_DIM*HW floats = 12544 B (< 320 KB/WGP -> high residency).
  __shared__ float smem[T_DIM * HW];

  const int c   = blockIdx.x;   // channel
  const int r   = blockIdx.y;   // roi
  const int tid = threadIdx.x;

  // --- bin geometry (exactly mirrors the JAX reference, f32 math) ---
  const float b0 = rois[r * 3 + 0];
  const float r1 = rois[r * 3 + 1];
  const float r2 = rois[r * 3 + 2];
  const int   bind      = (int)b0;
  const float roi_start = rintf(r1 * TSCALE);   // rint == round-half-even == jnp.round
  const float roi_end   = rintf(r2 * TSCALE);
  const float roi_len   = fmaxf(roi_end - roi_start + 1.0f, 1.0f);
  const float bs        = roi_len * 0.25f;      // exact: /POOLED

  int ls[POOLED], le[POOLED];
#pragma unroll
  for (int p = 0; p < POOLED; ++p) {
    float lsf = floorf((float)p * bs) + roi_start;
    float lef = ceilf((float)(p + 1) * bs) + roi_start;
    lsf = fminf(fmaxf(lsf, 0.0f), (float)T_DIM);
    lef = fminf(fmaxf(lef, 0.0f), (float)T_DIM);
    ls[p] = (int)lsf;
    le[p] = (int)lef;
  }
  // ls[] / le[] are nondecreasing in p, so the union window is [ls[0], le[3]).
  const int t0 = ls[0];
  const int t1 = le[POOLED - 1];
  const int nElems = (t1 - t0) * HW;   // <= 3136

  // --- stage window into LDS (async tensor-style copy on gfx1250) ---
  const float* gsrc = features + (size_t)bind * FEAT_NSTRIDE
                               + (size_t)c * FEAT_CSTRIDE
                               + (size_t)t0 * HW;
#if HAVE_ASYNC_LDS
  for (int i = tid; i < nElems; i += 256) {
    __builtin_amdgcn_global_load_async_to_lds_b32(
        (__attribute__((address_space(1))) int*)(gsrc + i),
        (__attribute__((address_space(3))) int*)(smem + i),
        /*offset=*/0, /*cpol=*/0);
  }
  wait_async0();                 // ASYNCcnt==0: all LDS writes landed
#else
  for (int i = tid; i < nElems; i += 256) smem[i] = gsrc[i];
#endif
  __syncthreads();

  // --- max-pool each (p, hw) from LDS; 196 contiguous output floats/block ---
  if (tid < POOLED * HW) {
    const int p  = tid / HW;
    const int hw = tid - p * HW;
    const int a = ls[p], b = le[p];
    float m = -INFINITY;
    for (int t = a; t < b; ++t)
      m = fmaxf(m, smem[(t - t0) * HW + hw]);
    if (b <= a) m = 0.0f;        // empty bin after clipping -> 0 (reference)
    out[(size_t)r * OUT_RSTRIDE + (size_t)c * OUT_CSTRIDE + tid] = m;
  }
}

extern "C" void kernel_launch(void* const* d_in, const int* in_sizes, int n_in,
                              void* d_out, int out_size, void* d_ws, size_t ws_size,
                              hipStream_t stream) {
  const float* features = (const float*)d_in[0];  // (2,512,64,7,7) f32
  const float* rois     = (const float*)d_in[1];  // (128,3) f32
  float*       out      = (float*)d_out;          // (128,512,4,7,7) f32

  dim3 grid(NCH, NUM_ROIS);   // one block per (channel, roi)
  roi_temporal_pool_kernel<<<grid, 256, 0, stream>>>(features, rois, out);
}